// LocationAwarePredictor_30081950941204
// MI455X (gfx1250) — compile-verified
//
#include <hip/hip_runtime.h>
#include <hip/hip_bf16.h>
#include <math.h>

// ---------------- problem constants ----------------
#define B_      8
#define N_      2000
#define H_      128
#define HEADS_  4
#define INF_    160          // H + H/4
#define DH_     32           // per-head out dim
#define BN_     8
#define HOR_    24
#define ALPHA_  0.2f
#define EPS_    1e-5f
#define NT_     (N_/16)      // 125 key/query tiles
#define RT_     ((B_*N_)/16) // 1000 row tiles of the fused [B*N, ...] GEMM
#define KST_    (INF_/4)     // 40 K-steps of 4 for the f32 WMMA
#define AW_     63           // ceil(N/32) packed adjacency words per row

typedef float v2f __attribute__((ext_vector_type(2)));
typedef float v8f __attribute__((ext_vector_type(8)));
typedef unsigned int v4u __attribute__((ext_vector_type(4)));
typedef int v4i __attribute__((ext_vector_type(4)));
typedef int v8i __attribute__((ext_vector_type(8)));

// D = A(16x4,f32) * B(4x16,f32) + C(16x16,f32) ; one matrix per wave32
__device__ __forceinline__ v8f wmma_f32_16x16x4(v2f a, v2f b, v8f c) {
  return __builtin_amdgcn_wmma_f32_16x16x4_f32(
      /*neg_a=*/false, a, /*neg_b=*/false, b,
      /*c_mod=*/(short)0, c, /*reuse_a=*/false, /*reuse_b=*/false);
}

__device__ __forceinline__ float wave_sum32(float v) {
#pragma unroll
  for (int off = 16; off > 0; off >>= 1) v += __shfl_xor(v, off, 32);
  return v;
}

// ---- Tensor Data Mover: async 2D tile (16 rows x 32 f32, row stride 128 f32)
// from global memory into LDS. Descriptor layout per CDNA5 ISA ch.8 (D#).
__device__ __forceinline__ void tdm_load_v_tile(unsigned lds_off, const float* gptr) {
  unsigned long long ga = (unsigned long long)(uintptr_t)gptr;
  v4u g0;
  g0[0] = 1u;                                             // count=1, user D#
  g0[1] = lds_off;                                        // lds_addr (bytes)
  g0[2] = (unsigned)(ga & 0xFFFFFFFFu);                   // global_addr[31:0]
  g0[3] = (unsigned)((ga >> 32) & 0x1FFFFFFu)             // global_addr[56:32]
        | (2u << 30);                                     // type = 2 ("image")
  v8i g1;
  g1[0] = (int)(2u << 16);      // workgroup_mask=0 (no cluster), data_size=4B
  g1[1] = (int)(32u << 16);     // atomic_barrier=0 | tensor_dim0[15:0] = 32
  g1[2] = (int)(16u << 16);     // tensor_dim0[31:16]=0 | tensor_dim1[15:0]=16
  g1[3] = (int)(32u << 16);     // tensor_dim1[31:16]=0 | tile_dim0 = 32
  g1[4] = 16;                   // tile_dim1 = 16, tile_dim2 = 0
  g1[5] = 128;                  // tensor_dim0_stride = 128 elements (low 32)
  g1[6] = 0;                    // stride hi | tensor_dim1_stride lo
  g1[7] = 0;
  v4i z4 = {0, 0, 0, 0};        // 2-D tile: D# groups 2/3 unused
#if __clang_major__ >= 23
  v8i z8 = {0, 0, 0, 0, 0, 0, 0, 0};
  __builtin_amdgcn_tensor_load_to_lds(g0, g1, z4, z4, z8, 0);
#else
  __builtin_amdgcn_tensor_load_to_lds(g0, g1, z4, z4, 0);
#endif
}

// ---------------- kernel 1: pack adjacency int32 -> bitmask ----------------
// 16 MB of int32 -> 504 KB of bits; the attention pass re-reads adj 32x, so
// this turns ~512 MB of HBM traffic into an L2-resident 500 KB structure.
__global__ void k_pack_adj(const int* __restrict__ adj, unsigned* __restrict__ adjp) {
  int idx = blockIdx.x * blockDim.x + threadIdx.x;  // row*AW_ + word
  if (idx >= N_ * AW_) return;
  int row = idx / AW_, w = idx % AW_;
  unsigned m = 0;
  int base = w * 32;
#pragma unroll 4
  for (int t = 0; t < 32; ++t) {
    int col = base + t;
    if (col < N_ && adj[row * N_ + col] > 0) m |= (1u << t);
  }
  adjp[idx] = m;
}

// ---------------- kernel 2: Wh = concat(x, loc_emb) @ W_cat  (WMMA) --------
// Treat the 4 heads as one [16000 x 160] x [160 x 128] GEMM where column
// block h*32..h*32+31 uses W_gat[h]. One wave per 16x16 output tile.
__global__ void __launch_bounds__(32)
k_wh_gemm(const float* __restrict__ x, const float* __restrict__ loc,
          const float* __restrict__ Wg, float* __restrict__ Wh) {
  const int lane = threadIdx.x;
  const int tile = blockIdx.x;        // 0..RT_*8-1
  const int ct   = tile & 7;          // output col tile (0..7) -> cols 16*ct..
  const int rt   = tile >> 3;         // output row tile (0..999)
  const int half = lane >> 4;         // K half-select for f32 WMMA A/B layout
  const int l16  = lane & 15;

  const int rowA = rt * 16 + l16;     // A-operand row owned by this lane
  const int nA   = rowA % N_;
  const size_t xbase = (size_t)rowA * H_;

  const int col  = ct * 16 + l16;     // output column (0..127)
  const int head = col >> 5;
  const int o    = col & 31;

  v8f acc = {};
#pragma unroll 4
  for (int kk = 0; kk < KST_; ++kk) {
    const int k0 = kk * 4 + half * 2;
    v2f a, b;
    // xl[row][k] = k<128 ? x[b,n,k] : loc_emb[n, k-128]
    a.x = (k0     < H_) ? x[xbase + k0]     : loc[nA * 32 + (k0 - H_)];
    a.y = (k0 + 1 < H_) ? x[xbase + k0 + 1] : loc[nA * 32 + (k0 + 1 - H_)];
    // W_cat[k][col] = W_gat[head][k][o]
    b.x = Wg[((size_t)head * INF_ + k0)     * DH_ + o];
    b.y = Wg[((size_t)head * INF_ + k0 + 1) * DH_ + o];
    acc = wmma_f32_16x16x4(a, b, acc);
  }
  // C/D layout: VGPR g -> row g (+8 for upper lane half), col = lane%16
#pragma unroll
  for (int g = 0; g < 8; ++g) {
    int row = rt * 16 + g + half * 8;
    Wh[(size_t)row * H_ + col] = acc[g];
  }
}

// ---------------- kernel 3: s_src / s_dst per (b,h,n) ----------------------
__global__ void k_attn_scores(const float* __restrict__ Wh,
                              const float* __restrict__ a_src,
                              const float* __restrict__ a_dst,
                              float* __restrict__ s_src, float* __restrict__ s_dst) {
  int idx = blockIdx.x * blockDim.x + threadIdx.x;  // (b*HEADS+h)*N + n
  if (idx >= B_ * HEADS_ * N_) return;
  int n  = idx % N_;
  int bh = idx / N_;
  int h  = bh % HEADS_;
  int bb = bh / HEADS_;
  const float* wrow = Wh + ((size_t)(bb * N_ + n)) * H_ + h * DH_;
  float ss = 0.f, sd = 0.f;
#pragma unroll 8
  for (int o = 0; o < DH_; ++o) {
    float w = wrow[o];
    ss = fmaf(w, a_src[h * DH_ + o], ss);
    sd = fmaf(w, a_dst[h * DH_ + o], sd);
  }
  s_src[idx] = ss;
  s_dst[idx] = sd;
}

// ---------------- kernel 4: flash-GAT attention (TDM + online softmax + WMMA)
// One wave per (b, h, 16-query tile). Streams 125 key tiles; never
// materializes the [B,4,N,N] score tensor. V tiles are DMA'd into LDS by the
// Tensor Data Mover while the wave does softmax VALU work; P.V runs on
// v_wmma_f32_16x16x4.
__global__ void __launch_bounds__(32)
k_attention(const float* __restrict__ Wh, const unsigned* __restrict__ adjp,
            const float* __restrict__ s_src, const float* __restrict__ s_dst,
            float* __restrict__ hprime) {
  __shared__ float Pt[16][17];     // score/prob tile (padded)
  __shared__ float Vt[16][32];     // key tile of Wh for this head (TDM dest)
  __shared__ float ssrc_t[16], sdst_t[16];
  __shared__ float rmax[16], rsum[16], rfac[16];

  const int lane = threadIdx.x;
  const int half = lane >> 4;
  const int l16  = lane & 15;

  const int rt = blockIdx.x % NT_;
  const int bh = blockIdx.x / NT_;
  const int h  = bh % HEADS_;
  const int bb = bh / HEADS_;

  const unsigned vt_lds = (unsigned)(uintptr_t)(&Vt[0][0]);

  if (lane < 16) {
    rmax[lane]   = -1e30f;
    rsum[lane]   = 0.f;
    ssrc_t[lane] = s_src[(size_t)bh * N_ + rt * 16 + lane];
  }
  __syncthreads();

  v8f acc0 = {}, acc1 = {};

  for (int ct = 0; ct < NT_; ++ct) {
    // kick off the TDM for this key tile: 16 rows x 32 f32, row stride 128 f32
    asm volatile("s_wait_dscnt 0x0" ::: "memory");  // prior Vt reads drained
    const float* vsrc = Wh + ((size_t)(bb * N_ + ct * 16)) * H_ + h * DH_;
    tdm_load_v_tile(vt_lds, vsrc);

    if (lane < 16) sdst_t[lane] = s_dst[(size_t)bh * N_ + ct * 16 + lane];
    __syncthreads();

    // raw scores while the DMA is in flight: additive GAT score + leakyrelu
    // + adjacency bitmask
#pragma unroll
    for (int t = 0; t < 8; ++t) {
      int e = lane + 32 * t;            // 0..255
      int i = e >> 4, j = e & 15;
      float s = ssrc_t[i] + sdst_t[j];
      s = (s > 0.f) ? s : ALPHA_ * s;
      int ni = rt * 16 + i, m = ct * 16 + j;
      unsigned bit = (adjp[(size_t)ni * AW_ + (m >> 5)] >> (m & 31)) & 1u;
      Pt[i][j] = bit ? s : -1e9f;
    }
    __syncthreads();

    // online-softmax row bookkeeping (lanes 0..15 own one row each)
    if (lane < 16) {
      float mx = rmax[lane];
#pragma unroll
      for (int j = 0; j < 16; ++j) mx = fmaxf(mx, Pt[lane][j]);
      float fac = __expf(rmax[lane] - mx);
      float sum = rsum[lane] * fac;
#pragma unroll
      for (int j = 0; j < 16; ++j) {
        float p = __expf(Pt[lane][j] - mx);
        Pt[lane][j] = p;
        sum += p;
      }
      rmax[lane] = mx; rsum[lane] = sum; rfac[lane] = fac;
    }
    __syncthreads();

    // rescale running accumulators by exp(m_old - m_new) per row
#pragma unroll
    for (int g = 0; g < 8; ++g) {
      float f = rfac[g + half * 8];
      acc0[g] *= f;
      acc1[g] *= f;
    }

    // V tile must have landed in LDS before the matrix ops consume it
    __builtin_amdgcn_s_wait_tensorcnt(0);

    // P[16x16] . V[16x32] : K=16 -> 4 WMMA K-steps, 2 output column halves
#pragma unroll
    for (int kk = 0; kk < 4; ++kk) {
      const int kb = kk * 4 + half * 2;
      v2f a, b0, b1;
      a.x  = Pt[l16][kb];      a.y  = Pt[l16][kb + 1];
      b0.x = Vt[kb][l16];      b0.y = Vt[kb + 1][l16];
      b1.x = Vt[kb][16 + l16]; b1.y = Vt[kb + 1][16 + l16];
      acc0 = wmma_f32_16x16x4(a, b0, acc0);
      acc1 = wmma_f32_16x16x4(a, b1, acc1);
    }
    __syncthreads();  // protect Pt/Vt before next tile overwrites them
  }

  // normalize by row sums and store h' (already in [B,N,H] head-major order)
#pragma unroll
  for (int g = 0; g < 8; ++g) {
    int i   = g + half * 8;
    float s = 1.0f / rsum[i];
    int n   = rt * 16 + i;
    size_t base = ((size_t)(bb * N_ + n)) * H_ + h * DH_;
    hprime[base + l16]      = acc0[g] * s;
    hprime[base + 16 + l16] = acc1[g] * s;
  }
}

// ---------------- kernel 5: LN + residual + MLP head + gated blend ---------
// One wave per node row; 8 rows per 256-thread block.
__global__ void __launch_bounds__(256)
k_epilogue(const float* __restrict__ hprime, const float* __restrict__ x,
           const float* __restrict__ last,
           const float* __restrict__ ln_gat_g, const float* __restrict__ ln_gat_b,
           const float* __restrict__ ln_g, const float* __restrict__ ln_b,
           const float* __restrict__ w_low, const float* __restrict__ b_low,
           const float* __restrict__ ln_bn_g, const float* __restrict__ ln_bn_b,
           const float* __restrict__ w_high, const float* __restrict__ b_high,
           const float* __restrict__ w_g1, const float* __restrict__ b_g1,
           const float* __restrict__ w_g2, const float* __restrict__ b_g2,
           float* __restrict__ out) {
  const int lane = threadIdx.x & 31;
  const int wave = threadIdx.x >> 5;
  const int row  = blockIdx.x * 8 + wave;       // 0..B*N-1
  if (row >= B_ * N_) return;

  const size_t base = (size_t)row * H_;
  float hp[4], res[4];
#pragma unroll
  for (int t = 0; t < 4; ++t) {
    hp[t]  = hprime[base + lane + 32 * t];
    res[t] = x[base + lane + 32 * t];
  }

  // LN(h') * g + b  (gat_out) then + residual
  float s = hp[0] + hp[1] + hp[2] + hp[3];
  float mean = wave_sum32(s) * (1.0f / H_);
  float vs = 0.f;
#pragma unroll
  for (int t = 0; t < 4; ++t) { float d = hp[t] - mean; vs += d * d; }
  float rstd = rsqrtf(wave_sum32(vs) * (1.0f / H_) + EPS_);
  float g1v[4];
#pragma unroll
  for (int t = 0; t < 4; ++t) {
    int f = lane + 32 * t;
    g1v[t] = (hp[t] - mean) * rstd * ln_gat_g[f] + ln_gat_b[f] + res[t];
  }

  // h = LN(gat_out + residual)
  s = g1v[0] + g1v[1] + g1v[2] + g1v[3];
  mean = wave_sum32(s) * (1.0f / H_);
  vs = 0.f;
#pragma unroll
  for (int t = 0; t < 4; ++t) { float d = g1v[t] - mean; vs += d * d; }
  rstd = rsqrtf(wave_sum32(vs) * (1.0f / H_) + EPS_);
  float hfin[4];
#pragma unroll
  for (int t = 0; t < 4; ++t) {
    int f = lane + 32 * t;
    hfin[t] = (g1v[t] - mean) * rstd * ln_g[f] + ln_b[f];
  }

  // x_low = h @ w_low + b_low  (128 -> 8), butterfly-reduced so every lane
  // holds all 8 values
  float xlo[BN_], gg[BN_];
#pragma unroll
  for (int j = 0; j < BN_; ++j) {
    float p = 0.f, q = 0.f;
#pragma unroll
    for (int t = 0; t < 4; ++t) {
      int f = lane + 32 * t;
      p = fmaf(hfin[t], w_low[f * BN_ + j], p);
      q = fmaf(hfin[t], w_g1[f * BN_ + j], q);
    }
    xlo[j] = wave_sum32(p) + b_low[j];
    gg[j]  = wave_sum32(q) + b_g1[j];
  }

  // x_mid = silu(LN(x_low)) over the 8-wide bottleneck (redundant per lane)
  float m8 = 0.f;
#pragma unroll
  for (int j = 0; j < BN_; ++j) m8 += xlo[j];
  m8 *= (1.0f / BN_);
  float v8 = 0.f;
#pragma unroll
  for (int j = 0; j < BN_; ++j) { float d = xlo[j] - m8; v8 += d * d; }
  float r8 = rsqrtf(v8 * (1.0f / BN_) + EPS_);
  float xmid[BN_], gsil[BN_];
#pragma unroll
  for (int j = 0; j < BN_; ++j) {
    float z = (xlo[j] - m8) * r8 * ln_bn_g[j] + ln_bn_b[j];
    xmid[j] = z / (1.0f + __expf(-z));            // silu
    float w = gg[j];
    gsil[j] = w / (1.0f + __expf(-w));            // silu
  }

  if (lane < HOR_) {
    float p = b_high[lane], q = b_g2[lane];
#pragma unroll
    for (int j = 0; j < BN_; ++j) {
      p = fmaf(xmid[j], w_high[j * HOR_ + lane], p);
      q = fmaf(gsil[j], w_g2[j * HOR_ + lane], q);
    }
    float gate = 1.0f / (1.0f + __expf(-q));
    float prog = last[row] * __expf(-0.1f * (float)(lane + 1));
    out[(size_t)row * HOR_ + lane] = gate * p + (1.0f - gate) * prog;
  }
}

// ---------------- host-side launch ----------------
extern "C" void kernel_launch(void* const* d_in, const int* in_sizes, int n_in,
                              void* d_out, int out_size, void* d_ws, size_t ws_size,
                              hipStream_t stream) {
  const float* x        = (const float*)d_in[0];
  const int*   adj      = (const int*)  d_in[1];
  const float* last     = (const float*)d_in[2];
  const float* loc      = (const float*)d_in[3];
  const float* Wg       = (const float*)d_in[4];
  const float* a_src    = (const float*)d_in[5];
  const float* a_dst    = (const float*)d_in[6];
  const float* ln_gat_g = (const float*)d_in[7];
  const float* ln_gat_b = (const float*)d_in[8];
  const float* ln_g     = (const float*)d_in[9];
  const float* ln_b     = (const float*)d_in[10];
  const float* w_low    = (const float*)d_in[11];
  const float* b_low    = (const float*)d_in[12];
  const float* ln_bn_g  = (const float*)d_in[13];
  const float* ln_bn_b  = (const float*)d_in[14];
  const float* w_high   = (const float*)d_in[15];
  const float* b_high   = (const float*)d_in[16];
  const float* w_g1     = (const float*)d_in[17];
  const float* b_g1     = (const float*)d_in[18];
  const float* w_g2     = (const float*)d_in[19];
  const float* b_g2     = (const float*)d_in[20];
  float* out = (float*)d_out;

  // workspace carve-up (all sizes 256B aligned)
  char* ws = (char*)d_ws;
  auto carve = [&](size_t bytes) {
    char* p = ws;
    ws += (bytes + 255) & ~(size_t)255;
    return p;
  };
  unsigned* adjp  = (unsigned*)carve((size_t)N_ * AW_ * sizeof(unsigned));
  float* Wh       = (float*)   carve((size_t)B_ * N_ * H_ * sizeof(float));
  float* s_src    = (float*)   carve((size_t)B_ * HEADS_ * N_ * sizeof(float));
  float* s_dst    = (float*)   carve((size_t)B_ * HEADS_ * N_ * sizeof(float));
  float* hprime   = (float*)   carve((size_t)B_ * N_ * H_ * sizeof(float));

  {
    int tot = N_ * AW_;
    k_pack_adj<<<(tot + 255) / 256, 256, 0, stream>>>(adj, adjp);
  }
  k_wh_gemm<<<RT_ * 8, 32, 0, stream>>>(x, loc, Wg, Wh);
  {
    int tot = B_ * HEADS_ * N_;
    k_attn_scores<<<(tot + 255) / 256, 256, 0, stream>>>(Wh, a_src, a_dst, s_src, s_dst);
  }
  k_attention<<<B_ * HEADS_ * NT_, 32, 0, stream>>>(Wh, adjp, s_src, s_dst, hprime);
  k_epilogue<<<(B_ * N_ + 7) / 8, 256, 0, stream>>>(
      hprime, x, last, ln_gat_g, ln_gat_b, ln_g, ln_b,
      w_low, b_low, ln_bn_g, ln_bn_b, w_high, b_high,
      w_g1, b_g1, w_g2, b_g2, out);
}